// ElmanRNN_7241314861502
// MI455X (gfx1250) — compile-verified
//
#include <hip/hip_runtime.h>
#include <math.h>

typedef __attribute__((ext_vector_type(2))) float v2f;
typedef __attribute__((ext_vector_type(8))) float v8f;

#define B_  64
#define S_  1024
#define I_  512
#define H_  512
#define LDSTR 514   // even stride: keeps v2f LDS loads 8B-aligned; stride%64==2 spreads banks
#define NWG2 32     // phase-2 workgroups (one per 16-wide H tile)

// ---------------------------------------------------------------------------
// Phase 1: out[r][n] = sum_k x[r][k] * Wih[n][k] + bih[n] + bhh[n]
// r = b*S + s (x_in is (B,S,I) row-major; out is (B,S,H) row-major)
// WG = 256 threads = 8 waves; each wave computes one 16x16 tile via
// chained V_WMMA_F32_16X16X4_F32; the 16xK slice of Wih lives in LDS.
// ---------------------------------------------------------------------------
__global__ __launch_bounds__(256) void rnn_xproj(const float* __restrict__ x,
                                                 const float* __restrict__ Wih,
                                                 const float* __restrict__ bih,
                                                 const float* __restrict__ bhh,
                                                 float* __restrict__ out) {
  __shared__ float ldsW[16 * LDSTR];
  const int col_tile  = blockIdx.x & 31;   // 32 H tiles
  const int row_group = blockIdx.x >> 5;   // 512 groups of 128 rows
  const int n0 = col_tile * 16;

  // cooperative load Wih[n0:n0+16][0:512] -> LDS
  for (int idx = threadIdx.x; idx < 16 * I_; idx += 256) {
    const int j = idx >> 9, k = idx & 511;
    ldsW[j * LDSTR + k] = Wih[(size_t)(n0 + j) * I_ + k];
  }
  __syncthreads();

  const int lane = threadIdx.x & 31;
  const int wave = threadIdx.x >> 5;
  const int r0   = (row_group * 8 + wave) * 16;
  const int jr   = lane & 15;            // row-within-tile for A/B fragments
  const int koff = (lane >> 4) << 1;     // K sub-offset: 0 (lanes 0-15) / 2 (16-31)
  const int mhi  = (lane >> 4) << 3;     // C fragment M offset: 0 / 8

  const float bias = bih[n0 + jr] + bhh[n0 + jr];
  v8f c0, c1;
  #pragma unroll
  for (int i = 0; i < 8; ++i) { c0[i] = bias; c1[i] = 0.0f; }

  const float* xrow = x + (size_t)(r0 + jr) * I_ + koff;
  const float* wrow = &ldsW[jr * LDSTR + koff];

  #pragma unroll 8
  for (int k0 = 0; k0 < I_; k0 += 8) {   // two independent WMMA chains
    v2f a0 = *(const v2f*)(xrow + k0);
    v2f b0 = *(const v2f*)(wrow + k0);
    c0 = __builtin_amdgcn_wmma_f32_16x16x4_f32(false, a0, false, b0, (short)0, c0, false, false);
    v2f a1 = *(const v2f*)(xrow + k0 + 4);
    v2f b1 = *(const v2f*)(wrow + k0 + 4);
    c1 = __builtin_amdgcn_wmma_f32_16x16x4_f32(false, a1, false, b1, (short)0, c1, false, false);
  }
  c0 = c0 + c1;

  #pragma unroll
  for (int r = 0; r < 8; ++r)
    out[(size_t)(r0 + r + mhi) * H_ + n0 + jr] = c0[r];
}

// ---------------------------------------------------------------------------
// Phase 2: persistent kernel, 32 WGs x 128 threads (4 waves).
// WG owns H tile [n0, n0+16); its Whh slice (16x512, ~32 KB) stays in LDS for
// the whole sequence. Wave w owns batch tile [16w, 16w+16).
// Per step t: C = xp fragment (read from out[:,t,:]); if t>0 accumulate
// h_{t-1} @ Whh^T via WMMA (h read in-place from out[:,t-1,:]); tanh; store.
// Device-wide barrier (release/acquire atomic counter + s_sleep spin) per
// step; next step's xp fragment is prefetched before the barrier.
// ---------------------------------------------------------------------------
__global__ __launch_bounds__(128) void rnn_recur(const float* __restrict__ Whh,
                                                 float* __restrict__ out,
                                                 unsigned* __restrict__ cnt) {
  __shared__ float ldsW[16 * LDSTR];
  const int n0 = blockIdx.x * 16;

  for (int idx = threadIdx.x; idx < 16 * H_; idx += 128) {
    const int j = idx >> 9, k = idx & 511;
    ldsW[j * LDSTR + k] = Whh[(size_t)(n0 + j) * H_ + k];
  }
  __syncthreads();

  const int lane = threadIdx.x & 31;
  const int wave = threadIdx.x >> 5;
  const int b0   = wave * 16;
  const int jr   = lane & 15;
  const int koff = (lane >> 4) << 1;
  const int mhi  = (lane >> 4) << 3;
  const float* wrow = &ldsW[jr * LDSTR + koff];

  for (int t = 0; t < S_; ++t) {
    // C init = xp fragment for this step
    v8f c0, c1;
    #pragma unroll
    for (int r = 0; r < 8; ++r) {
      c0[r] = out[((size_t)(b0 + r + mhi) * S_ + t) * H_ + n0 + jr];
      c1[r] = 0.0f;
    }

    if (t > 0) {  // accumulate h_{t-1} @ Whh^T (h_0 == 0)
      const float* hrow = out + ((size_t)(b0 + jr) * S_ + (t - 1)) * H_ + koff;
      #pragma unroll 8
      for (int k0 = 0; k0 < H_; k0 += 8) {
        v2f a0 = *(const v2f*)(hrow + k0);
        v2f b0v = *(const v2f*)(wrow + k0);
        c0 = __builtin_amdgcn_wmma_f32_16x16x4_f32(false, a0, false, b0v, (short)0, c0, false, false);
        v2f a1 = *(const v2f*)(hrow + k0 + 4);
        v2f b1v = *(const v2f*)(wrow + k0 + 4);
        c1 = __builtin_amdgcn_wmma_f32_16x16x4_f32(false, a1, false, b1v, (short)0, c1, false, false);
      }
    }
    c0 = c0 + c1;

    #pragma unroll
    for (int r = 0; r < 8; ++r)
      out[((size_t)(b0 + r + mhi) * S_ + t) * H_ + n0 + jr] = tanhf(c0[r]);

    if (t + 1 < S_) {
      // pull next step's xp fragment toward the caches while we wait
      #pragma unroll
      for (int r = 0; r < 8; ++r)
        __builtin_prefetch(&out[((size_t)(b0 + r + mhi) * S_ + (t + 1)) * H_ + n0 + jr], 0, 1);

      // device-wide barrier: make h_t visible to all WGs
      __threadfence();
      __syncthreads();
      if (threadIdx.x == 0) {
        __hip_atomic_fetch_add(cnt, 1u, __ATOMIC_RELEASE, __HIP_MEMORY_SCOPE_AGENT);
        const unsigned target = (unsigned)(t + 1) * (unsigned)NWG2;
        while (__hip_atomic_load(cnt, __ATOMIC_ACQUIRE, __HIP_MEMORY_SCOPE_AGENT) < target)
          __builtin_amdgcn_s_sleep(2);
      }
      __syncthreads();
    }
  }
}

extern "C" void kernel_launch(void* const* d_in, const int* in_sizes, int n_in,
                              void* d_out, int out_size, void* d_ws, size_t ws_size,
                              hipStream_t stream) {
  const float* x   = (const float*)d_in[0];   // (B,S,I)
  const float* Wih = (const float*)d_in[1];   // (H,I)
  const float* Whh = (const float*)d_in[2];   // (H,H)
  const float* bih = (const float*)d_in[3];   // (H)
  const float* bhh = (const float*)d_in[4];   // (H)
  float* out = (float*)d_out;                 // (B,S,H)
  unsigned* cnt = (unsigned*)d_ws;

  hipMemsetAsync(d_ws, 0, 256, stream);       // zero barrier counter (capture-safe)

  // Phase 1: 65536 rows / 128 per WG = 512 row groups x 32 col tiles
  rnn_xproj<<<dim3(512 * 32), 256, 0, stream>>>(x, Wih, bih, bhh, out);
  // Phase 2: persistent recurrence
  rnn_recur<<<dim3(NWG2), 128, 0, stream>>>(Whh, out, cnt);
}